// MeanEmbedding_12008728559640
// MI455X (gfx1250) — compile-verified
//
#include <hip/hip_runtime.h>
#include <hip/hip_bf16.h>

typedef __attribute__((ext_vector_type(2))) float v2f;
typedef __attribute__((ext_vector_type(8))) float v8f;

#define S_LEN 4096
#define D_DIM 512

// One block = one (batch, 16-column tile). 8 waves interleave over S in
// 4-row chunks (stride 32); each chunk is reduced with V_WMMA_F32_16X16X4_F32:
//   A = mask (rows replicated over M), B = 4x16 tile of xs, C accumulates.
// Row M=0 of C (VGPR0, lanes 0-15) then holds the 16 column sums per wave.
__global__ __launch_bounds__(256) void MeanEmbedding_wmma_kernel(
    const float* __restrict__ xs, const long long* __restrict__ xs_len,
    float* __restrict__ out)
{
    const int dtile = blockIdx.x;          // 0..31  (16 columns each)
    const int b     = blockIdx.y;          // 0..15
    const int tid   = threadIdx.x;
    const int lane  = tid & 31;
    // Force wave-uniform scalars so the S-loop lowers to scalar branches
    // (keeps EXEC all-ones around WMMA, no per-lane loop mask).
    const int wave  = __builtin_amdgcn_readfirstlane(tid >> 5);     // 0..7
    const int len   = __builtin_amdgcn_readfirstlane((int)xs_len[b]);

    const float* __restrict__ xsb = xs + (size_t)b * (S_LEN * D_DIM);

    const int col   = lane & 15;           // B-matrix column N
    const int kbase = (lane >> 4) << 1;    // K=0,1 in lanes 0-15; K=2,3 in 16-31
    const int dcol  = (dtile << 4) + col;

    v8f c = {};

    // One 4-row x 16-col chunk: 2 non-temporal loads + 1 fp32 WMMA.
    #define CHUNK(S0)                                                         \
    {                                                                         \
        const int r0 = (S0) + kbase;                                          \
        v2f a, bm;                                                            \
        a.x = (r0     < len) ? 1.0f : 0.0f;   /* mask folds ragged tail */    \
        a.y = (r0 + 1 < len) ? 1.0f : 0.0f;                                   \
        const float* p = xsb + (size_t)r0 * D_DIM + dcol;                     \
        bm.x = __builtin_nontemporal_load(p);          /* row r0   */         \
        bm.y = __builtin_nontemporal_load(p + D_DIM);  /* row r0+1 */         \
        c = __builtin_amdgcn_wmma_f32_16x16x4_f32(                            \
                /*neg_a=*/false, a, /*neg_b=*/false, bm,                      \
                /*c_mod=*/(short)0, c, /*reuse_a=*/false, /*reuse_b=*/false); \
    }

    // Main loop unrolled x2 (4 loads in flight per wave before the waits).
    // Both chunks satisfy s < len, so all row reads stay inside the batch.
    int s0 = wave << 2;
    for (; s0 + 32 < len; s0 += 64) {
        CHUNK(s0)
        CHUNK(s0 + 32)
    }
    // Uniform tail: at most one remaining chunk for this wave.
    if (s0 < len) {
        CHUNK(s0)
    }
    #undef CHUNK

    // Cross-wave reduction of the 8 partial sums (row M=0 of each wave's C).
    __shared__ float lds[8][16];
    if (lane < 16) lds[wave][lane] = c[0];
    __syncthreads();

    if (tid < 16) {
        float s = 0.0f;
        #pragma unroll
        for (int w = 0; w < 8; ++w) s += lds[w][tid];
        out[(size_t)b * D_DIM + (dtile << 4) + tid] = s / (float)len;
    }
}

extern "C" void kernel_launch(void* const* d_in, const int* in_sizes, int n_in,
                              void* d_out, int out_size, void* d_ws, size_t ws_size,
                              hipStream_t stream) {
    const float*     xs     = (const float*)d_in[0];
    const long long* xs_len = (const long long*)d_in[1];  // int64 per reference
    float*           out    = (float*)d_out;              // [16, 512] fp32

    dim3 grid(D_DIM / 16, 16);   // 32 column tiles x 16 batches = 512 blocks
    MeanEmbedding_wmma_kernel<<<grid, 256, 0, stream>>>(xs, xs_len, out);
}